// Equalize_49082886259136
// MI455X (gfx1250) — compile-verified
//
#include <hip/hip_runtime.h>

typedef __attribute__((ext_vector_type(2))) float v2f;
typedef __attribute__((ext_vector_type(8))) float v8f;

#define NBINS 256
#define B_DIM 32
#define C_DIM 3
#define BC (B_DIM * C_DIM)          // 96 channels
#define HW (1024 * 1024)            // pixels per channel
#define BLOCKS_PER_CHAN 64
#define PX_PER_BLOCK (HW / BLOCKS_PER_CHAN)   // 16384
#define WAVES_PER_BLOCK 8

// ---------------------------------------------------------------------------
// Kernel 1: per-channel 256-bin histogram.
// 8 per-wave LDS sub-histograms (ds_add_u32), merged once into global bins.
// ---------------------------------------------------------------------------
__global__ void eq_hist_kernel(const float* __restrict__ img,
                               int* __restrict__ ghist) {
    __shared__ int lhist[WAVES_PER_BLOCK][NBINS];
    const int tid  = threadIdx.x;
    const int wave = tid >> 5;

    for (int i = tid; i < WAVES_PER_BLOCK * NBINS; i += blockDim.x)
        ((int*)lhist)[i] = 0;
    __syncthreads();

    const int chan = blockIdx.x / BLOCKS_PER_CHAN;
    const int blk  = blockIdx.x % BLOCKS_PER_CHAN;
    const float4* vin =
        (const float4*)(img + (size_t)chan * HW + (size_t)blk * PX_PER_BLOCK);
    const int n4 = PX_PER_BLOCK / 4;

    for (int i = tid; i < n4; i += blockDim.x) {
        __builtin_prefetch((const void*)(vin + i + blockDim.x), 0, 0);
        float4 p = vin[i];
        atomicAdd(&lhist[wave][(int)p.x], 1);
        atomicAdd(&lhist[wave][(int)p.y], 1);
        atomicAdd(&lhist[wave][(int)p.z], 1);
        atomicAdd(&lhist[wave][(int)p.w], 1);
    }
    __syncthreads();

    for (int b = tid; b < NBINS; b += blockDim.x) {
        int s = 0;
        #pragma unroll
        for (int w = 0; w < WAVES_PER_BLOCK; ++w) s += lhist[w][b];
        atomicAdd(&ghist[chan * NBINS + b], s);
    }
}

// ---------------------------------------------------------------------------
// Kernel 2: per-channel LUT. One wave (32 lanes) per channel.
// 256-bin inclusive cumsum via WMMA: hist viewed as M[16][16];
// P = M * T (T upper-triangular ones) with 4 chained v_wmma_f32_16x16x4_f32,
// then add exclusive prefix of row sums (column 15 of P).
// ---------------------------------------------------------------------------
__global__ void eq_lut_kernel(const int* __restrict__ ghist,
                              int* __restrict__ glut) {
    __shared__ float fh[NBINS];
    __shared__ int   ih[NBINS];
    __shared__ float rowoff[16];
    __shared__ int   icum[NBINS];

    const int lane = threadIdx.x;          // 0..31, one full wave
    const int bc   = blockIdx.x;
    const int* h   = ghist + bc * NBINS;

    int total = 0, lastIdx = -1;
    #pragma unroll
    for (int k = 0; k < 8; ++k) {
        int i = lane + 32 * k;             // increasing i per lane
        int v = h[i];
        ih[i] = v;
        fh[i] = (float)v;
        total += v;
        if (v != 0) lastIdx = i;
    }
    __syncthreads();

    // wave32 reductions: total count, last nonzero bin index
    #pragma unroll
    for (int off = 16; off > 0; off >>= 1) {
        total += __shfl_xor(total, off, 32);
        int o = __shfl_xor(lastIdx, off, 32);
        lastIdx = lastIdx > o ? lastIdx : o;
    }
    const int lastVal = ih[lastIdx < 0 ? 0 : lastIdx];
    const int step    = (total - lastVal) / (NBINS - 1);

    // ---- WMMA: P = M * T, chained along K in 16x16x4 steps ----
    const int half = lane >> 4;            // 0: lanes 0-15, 1: lanes 16-31
    const int l    = lane & 15;
    v8f acc = {};
    #pragma unroll
    for (int kk = 0; kk < 4; ++kk) {
        // A-layout (f32 16x4): lanes 0-15 M=l hold K=4kk+0/+1,
        //                      lanes 16-31 M=l hold K=4kk+2/+3
        const int k0 = 4 * kk + 2 * half;
        v2f a, b;
        a.x = fh[l * 16 + k0];
        a.y = fh[l * 16 + k0 + 1];
        // B-layout (f32 4x16): T[k][n] = (k <= n), N = l
        b.x = (k0     <= l) ? 1.0f : 0.0f;
        b.y = (k0 + 1 <= l) ? 1.0f : 0.0f;
        acc = __builtin_amdgcn_wmma_f32_16x16x4_f32(
            false, a, false, b, (short)0, acc, false, false);
    }

    // D layout: acc[j] = P(m = j + 8*half, n = l). Row sums live at n == 15.
    if (l == 15) {
        #pragma unroll
        for (int j = 0; j < 8; ++j) rowoff[8 * half + j] = acc[j];
    }
    __syncthreads();
    if (lane == 0) {
        float run = 0.0f;
        for (int m = 0; m < 16; ++m) { float t = rowoff[m]; rowoff[m] = run; run += t; }
    }
    __syncthreads();

    #pragma unroll
    for (int j = 0; j < 8; ++j) {
        const int m = j + 8 * half;
        const float c = acc[j] + rowoff[m];      // inclusive cumsum, exact in f32
        icum[m * 16 + l] = (int)(c + 0.5f);
    }
    __syncthreads();

    // LUT: lut[0]=0; lut[i] = clamp((cum[i-1] + step/2) / step); identity if step==0
    #pragma unroll
    for (int k = 0; k < 8; ++k) {
        const int i = lane + 32 * k;
        int lut;
        if (step == 0)      lut = i;             // channel returned unchanged
        else if (i == 0)    lut = 0;
        else {
            lut = (icum[i - 1] + (step >> 1)) / step;
            lut = lut < 0 ? 0 : (lut > 255 ? 255 : lut);
        }
        glut[bc * NBINS + i] = lut;
    }
}

// ---------------------------------------------------------------------------
// Kernel 3: remap via per-channel LUT staged in LDS. Pure streaming b128.
// ---------------------------------------------------------------------------
__global__ void eq_remap_kernel(const float* __restrict__ img,
                                const int* __restrict__ glut,
                                float* __restrict__ out) {
    __shared__ float flut[NBINS];
    const int tid  = threadIdx.x;
    const int chan = blockIdx.x / BLOCKS_PER_CHAN;
    const int blk  = blockIdx.x % BLOCKS_PER_CHAN;

    for (int i = tid; i < NBINS; i += blockDim.x)
        flut[i] = (float)glut[chan * NBINS + i];
    __syncthreads();

    const size_t base = (size_t)chan * HW + (size_t)blk * PX_PER_BLOCK;
    const float4* vin = (const float4*)(img + base);
    float4*       vout = (float4*)(out + base);
    const int n4 = PX_PER_BLOCK / 4;

    for (int i = tid; i < n4; i += blockDim.x) {
        __builtin_prefetch((const void*)(vin + i + blockDim.x), 0, 0);
        float4 p = vin[i];
        float4 q;
        q.x = flut[(int)p.x];
        q.y = flut[(int)p.y];
        q.z = flut[(int)p.z];
        q.w = flut[(int)p.w];
        vout[i] = q;
    }
}

extern "C" void kernel_launch(void* const* d_in, const int* in_sizes, int n_in,
                              void* d_out, int out_size, void* d_ws, size_t ws_size,
                              hipStream_t stream) {
    (void)in_sizes; (void)n_in; (void)out_size; (void)ws_size;
    const float* img = (const float*)d_in[0];
    float* out = (float*)d_out;

    int* ghist = (int*)d_ws;                 // BC*NBINS ints
    int* glut  = ghist + BC * NBINS;         // BC*NBINS ints

    hipMemsetAsync(ghist, 0, (size_t)BC * NBINS * sizeof(int), stream);

    eq_hist_kernel<<<dim3(BC * BLOCKS_PER_CHAN), 256, 0, stream>>>(img, ghist);
    eq_lut_kernel<<<dim3(BC), 32, 0, stream>>>(ghist, glut);
    eq_remap_kernel<<<dim3(BC * BLOCKS_PER_CHAN), 256, 0, stream>>>(img, glut, out);
}